// CustomRationalLayer_45243185496814
// MI455X (gfx1250) — compile-verified
//
#include <hip/hip_runtime.h>

typedef float v2f __attribute__((ext_vector_type(2)));
typedef float v8f __attribute__((ext_vector_type(8)));

#define B_TOT 256
#define I_TOT 512
#define O_TOT 512
#define CH    32          // i-chunk staged in LDS per block iteration

// Block = 256 threads = 8 waves. One b-tile (16 batch rows) per block, one
// o-tile (16 output cols) per wave. Grid = (O/16/8, B/16) = (4, 16).
__global__ __launch_bounds__(256)
void rational_wmma_kernel(const float* __restrict__ x,
                          const float* __restrict__ tanh_range,
                          const float* __restrict__ mole_coef,
                          const float* __restrict__ deno_coef,
                          float* __restrict__ out)
{
    __shared__ float t_lds[CH * 16];
    __shared__ float ax_lds[CH * 16];

    const int tid  = threadIdx.x;
    const int wave = tid >> 5;
    const int lane = tid & 31;
    const int half = lane >> 4;      // 0: lanes 0-15, 1: lanes 16-31
    const int row  = lane & 15;      // A-matrix M row (o) / B-matrix N col (b)

    const int b_tile = blockIdx.y;                 // 0..15
    const int o_tile = blockIdx.x * 8 + wave;      // 0..31
    const int b0     = b_tile * 16;
    const int o      = o_tile * 16 + row;          // this lane's A row (o index)

    const float r = tanh_range[0];

    const float* mc = mole_coef + (size_t)o * I_TOT * 6;   // +i*6+p
    const float* dc = deno_coef + (size_t)o * I_TOT * 4;   // +i*4+q

    v8f acc = {};

    for (int i0 = 0; i0 < I_TOT; i0 += CH) {
        __syncthreads();
        // Stage x chunk: coalesced loads, tanh + |x| precomputed once for all
        // 8 waves. LDS layout [ii][16] -> 16-lane column reads conflict-free.
        for (int idx = tid; idx < 16 * CH; idx += 256) {
            const int rr = idx / CH;                // b row within tile
            const int cc = idx % CH;                // i within chunk
            const float xv = x[(size_t)(b0 + rr) * I_TOT + i0 + cc];
            t_lds [cc * 16 + rr] = tanhf(xv * r);
            ax_lds[cc * 16 + rr] = __builtin_fabsf(xv);
        }
        __syncthreads();

        #pragma unroll 2
        for (int ii = 0; ii < CH; ++ii) {
            const int i = i0 + ii;
            const float t  = t_lds [ii * 16 + row];   // per-lane (N = b)
            const float ax = ax_lds[ii * 16 + row];
            const float t2 = t * t, t3 = t2 * t, t4 = t2 * t2, t5 = t4 * t;

            // ---- A matrices: coefficients, M = o, K = vgpr + 2*half ----
            const float* mci = mc + i * 6;
            const float* dci = dc + i * 4;
            v2f am1 = *(const v2f*)(mci + 2 * half);   // p = 2h, 2h+1
            v2f am2 = *(const v2f*)(mci + 4);          // p4, p5 (zero for K=2,3)
            am2.x = half ? 0.0f : am2.x;
            am2.y = half ? 0.0f : am2.y;
            v2f ad  = *(const v2f*)(dci + 2 * half);   // powers 1+2h, 2+2h

            // ---- B matrices: t powers, N = b (per-lane), K = vgpr + 2*half ----
            v2f bm1, bm2, bd;
            bm1.x = half ? t2 : 1.0f;  bm1.y = half ? t3 : t;
            bm2.x = half ? 0.0f : t4;  bm2.y = half ? 0.0f : t5;
            bd.x  = half ? t3 : t;     bd.y  = half ? t4 : t2;

            v8f zero = {};
            v8f m = __builtin_amdgcn_wmma_f32_16x16x4_f32(
                        false, am1, false, bm1, (short)0, zero, false, false);
            m     = __builtin_amdgcn_wmma_f32_16x16x4_f32(
                        false, am2, false, bm2, (short)0, m, false, false);
            v8f d = __builtin_amdgcn_wmma_f32_16x16x4_f32(
                        false, ad,  false, bd,  (short)0, zero, false, false);

            // Rational part: acc += m / (|d|*|x| + 1). ax is per-lane scalar.
            #pragma unroll
            for (int e = 0; e < 8; ++e) {
                const float denom = __builtin_fmaf(__builtin_fabsf(d[e]), ax, 1.0f);
                acc[e] += m[e] * __builtin_amdgcn_rcpf(denom);
            }
        }
    }

    // D layout: element e -> o column (o_tile*16 + e + 8*half), N = row -> b.
    // 8 consecutive columns per lane -> two 128-bit stores.
    float* op = out + (size_t)(b0 + row) * O_TOT + o_tile * 16 + half * 8;
    *(float4*)(op)     = make_float4(acc[0], acc[1], acc[2], acc[3]);
    *(float4*)(op + 4) = make_float4(acc[4], acc[5], acc[6], acc[7]);
}

extern "C" void kernel_launch(void* const* d_in, const int* in_sizes, int n_in,
                              void* d_out, int out_size, void* d_ws, size_t ws_size,
                              hipStream_t stream) {
    const float* x  = (const float*)d_in[0];
    const float* tr = (const float*)d_in[1];
    const float* mc = (const float*)d_in[2];
    const float* dc = (const float*)d_in[3];
    float* out = (float*)d_out;

    dim3 grid(O_TOT / 16 / 8, B_TOT / 16);   // (4, 16)
    rational_wmma_kernel<<<grid, 256, 0, stream>>>(x, tr, mc, dc, out);
}